// EdgeAttention_80573586473089
// MI455X (gfx1250) — compile-verified
//
#include <hip/hip_runtime.h>
#include <cstdint>

// Problem constants (fixed by setup_inputs): B=16, C=256, H=W=128, NCHW fp32.
#define BATCH      16
#define CHANS      256
#define IMG_H      128
#define IMG_W      128
#define PLANE      (IMG_H * IMG_W)

#define TILE_ROWS  32              // output rows per workgroup
#define STRIPES    (IMG_H / TILE_ROWS)   // 4
#define XS_ROWS    (TILE_ROWS + 4) // 36: input rows needed (2-halo each side)
#define XS_W       136             // 4 left pad + 128 + 2 right halo + align; row = 544B
#define ES_ROWS    (TILE_ROWS + 2) // 34: edge rows needed (1-halo each side)
#define ES_W       132             // 1 left pad + 128 + right pad + align

typedef float v4f __attribute__((ext_vector_type(4)));
typedef int   v4i __attribute__((ext_vector_type(4)));
typedef __attribute__((address_space(1))) v4i GV4;   // global int4
typedef __attribute__((address_space(3))) v4i LV4;   // LDS int4

// --- CDNA5 async global->LDS copy (GLOBAL_LOAD_ASYNC_TO_LDS_B128, ASYNCcnt) ---
__device__ __forceinline__ void async_copy_b128(const float* gsrc, float* ldst) {
#if __has_builtin(__builtin_amdgcn_global_load_async_to_lds_b128)
    __builtin_amdgcn_global_load_async_to_lds_b128(
        (GV4*)(void*)gsrc, (LV4*)(void*)ldst, 0, 0);
#else
    asm volatile("global_load_async_to_lds_b128 %0, %1, off"
                 :
                 : "v"((LV4*)(void*)ldst), "v"((GV4*)(void*)gsrc)
                 : "memory");
#endif
}

__device__ __forceinline__ void wait_async0() {
#if __has_builtin(__builtin_amdgcn_s_wait_asynccnt)
    __builtin_amdgcn_s_wait_asynccnt(0);
#else
    asm volatile("s_wait_asynccnt 0" ::: "memory");
#endif
}

__global__ __launch_bounds__(256) void EdgeAttention_fused_kernel(
    const float* __restrict__ x,        // [B,C,H,W]
    const float* __restrict__ dww,      // [C,1,3,3]
    const float* __restrict__ bn_gamma, // [C]
    const float* __restrict__ bn_beta,  // [C]
    const float* __restrict__ bn_mean,  // [C]
    const float* __restrict__ bn_var,   // [C]
    float* __restrict__ out)            // [B,C,H,W]
{
    __shared__ __align__(16) float xs[XS_ROWS * XS_W]; // input tile (zero halo)
    __shared__ __align__(16) float es[ES_ROWS * ES_W]; // edge tile  (zero halo)

    const int t      = threadIdx.x;
    const int blk    = blockIdx.x;          // [0, B*C*STRIPES)
    const int stripe = blk & (STRIPES - 1);
    const int plane  = blk >> 2;            // b*C + c
    const int c      = plane & (CHANS - 1);
    const int r0     = stripe * TILE_ROWS;

    const float* xplane = x   + (size_t)plane * PLANE;
    float*       oplane = out + (size_t)plane * PLANE + (size_t)r0 * IMG_W;

    // ------------- phase A: issue async copies of valid input rows FIRST ----------
    // need global rows [r0-2, r0+33] clipped to [0, 127]; x image placed at col 4.
    // async engine writes ONLY interior cells (valid rows x cols 4..131).
    const int grLo = (r0 - 2 < 0) ? 0 : (r0 - 2);
    const int grHi = (r0 + XS_ROWS - 3 > IMG_H - 1) ? (IMG_H - 1) : (r0 + XS_ROWS - 3);
    const int nChunks = (grHi - grLo + 1) * (IMG_W / 4); // 16B chunks
    for (int i = t; i < nChunks; i += 256) {
        const int rr = i >> 5;            // 32 chunks per row
        const int cc = (i & 31) << 2;     // starting column (floats)
        const int gr = grLo + rr;
        const int lr = gr - (r0 - 2);
        async_copy_b128(xplane + gr * IMG_W + cc, &xs[lr * XS_W + 4 + cc]);
    }

    // ------------- phase B: zero ONLY halo cells while async loads stream ---------
    {
        const v4f z4 = (v4f)0.f;
        // xs side columns 0..3 and 132..135, every row (covers read cols 3 / 132)
        for (int i = t; i < XS_ROWS * 2; i += 256) {
            const int r = i >> 1;
            *(v4f*)&xs[r * XS_W + ((i & 1) ? 132 : 0)] = z4;
        }
        // xs out-of-image rows (gr < 0 or gr > 127): full-row zero
        const int nInvTop = (r0 == 0) ? 2 : 0;                    // lr 0,1
        const int nInvBot = (r0 == IMG_H - TILE_ROWS) ? 2 : 0;    // lr 34,35
        for (int i = t; i < (nInvTop + nInvBot) * (XS_W / 4); i += 256) {
            const int rI = i / (XS_W / 4);
            const int cc = (i % (XS_W / 4)) * 4;
            const int lr = (rI < nInvTop) ? rI : (XS_ROWS - nInvBot + (rI - nInvTop));
            *(v4f*)&xs[lr * XS_W + cc] = z4;
        }
        // es padding columns: edge col -1 (idx 0) and edge col 128 (idx 129), all rows
        for (int i = t; i < ES_ROWS * 2; i += 256) {
            const int r = i >> 1;
            es[r * ES_W + ((i & 1) ? 129 : 0)] = 0.f;
        }
        // es padding rows: ge = -1 (le 0, stripe 0) / ge = 128 (le 33, last stripe)
        const int neTop = (r0 == 0) ? 1 : 0;
        const int neBot = (r0 == IMG_H - TILE_ROWS) ? 1 : 0;
        for (int i = t; i < (neTop + neBot) * ES_W; i += 256) {
            const int rI = i / ES_W;
            const int cc = i % ES_W;
            const int le = (rI < neTop) ? 0 : (ES_ROWS - 1);
            es[le * ES_W + cc] = 0.f;
        }
    }

    // single join: ASYNCcnt for the DMA fill, DScnt (implicit in barrier) for halos
    wait_async0();
    __syncthreads();

    // ------------- phase 2: Sobel -> edge = 0.5*(|gx|+|gy|) into LDS --------------
    // edge rows needed: ge in [r0-1, r0+32] clipped to [0,127]; es row le = ge-(r0-1).
    const int geLo = (r0 - 1 < 0) ? 0 : (r0 - 1);
    const int geHi = (r0 + TILE_ROWS > IMG_H - 1) ? (IMG_H - 1) : (r0 + TILE_ROWS);
    const int nEdge = (geHi - geLo + 1) * IMG_W;
    for (int i = t; i < nEdge; i += 256) {
        const int er = i >> 7;            // /128
        const int ec = i & 127;
        const int ge = geLo + er;
        const int le = ge - (r0 - 1);
        const float* x0 = &xs[(le + 0) * XS_W + 4 + ec]; // row ge-1 (zeros if OOB)
        const float* x1 = &xs[(le + 1) * XS_W + 4 + ec]; // row ge
        const float* x2 = &xs[(le + 2) * XS_W + 4 + ec]; // row ge+1
        const float a = x0[-1], b = x0[0], cc2 = x0[1];
        const float d = x1[-1],             e = x1[1];
        const float f = x2[-1], g = x2[0], h = x2[1];
        const float gx = (cc2 - a) + 2.f * (e - d) + (h - f);
        const float gy = (f   - a) + 2.f * (g - b) + (h - cc2);
        es[le * ES_W + 1 + ec] = 0.5f * (fabsf(gx) + fabsf(gy));
    }
    __syncthreads();

    // ------------- phase 3: depthwise 3x3 + BN + ReLU, 4x4 microtile/thread -------
    const float* wk  = dww + c * 9;       // uniform per block -> scalar loads
    const float w00 = wk[0], w01 = wk[1], w02 = wk[2];
    const float w10 = wk[3], w11 = wk[4], w12 = wk[5];
    const float w20 = wk[6], w21 = wk[7], w22 = wk[8];
    const float inv  = bn_gamma[c] / sqrtf(bn_var[c] + 1e-5f);
    const float bias = bn_beta[c] - bn_mean[c] * inv;

    const int w4 = (t & 31) * 4;          // output col base
    const int rg = t >> 5;                // 0..7 -> 4 rows each
    #pragma unroll
    for (int rr = 0; rr < 4; ++rr) {
        const int orow = rg * 4 + rr;     // 0..31 within stripe
        // output row h=r0+orow reads edge rows le=orow..orow+2, cols (1+w-1 .. 1+w+1)
        const float* e0 = &es[(orow + 0) * ES_W + 1 + w4];
        const float* e1 = &es[(orow + 1) * ES_W + 1 + w4];
        const float* e2 = &es[(orow + 2) * ES_W + 1 + w4];
        v4f o;
        #pragma unroll
        for (int j = 0; j < 4; ++j) {
            float s = w00 * e0[j - 1] + w01 * e0[j] + w02 * e0[j + 1]
                    + w10 * e1[j - 1] + w11 * e1[j] + w12 * e1[j + 1]
                    + w20 * e2[j - 1] + w21 * e2[j] + w22 * e2[j + 1];
            s = s * inv + bias;
            o[j] = s > 0.f ? s : 0.f;
        }
        __builtin_nontemporal_store(o, (v4f*)(oplane + orow * IMG_W + w4));
    }
}

extern "C" void kernel_launch(void* const* d_in, const int* in_sizes, int n_in,
                              void* d_out, int out_size, void* d_ws, size_t ws_size,
                              hipStream_t stream) {
    const float* x     = (const float*)d_in[0];
    const float* dww   = (const float*)d_in[1];
    const float* gamma = (const float*)d_in[2];
    const float* beta  = (const float*)d_in[3];
    const float* mean  = (const float*)d_in[4];
    const float* var   = (const float*)d_in[5];
    float* out = (float*)d_out;

    const int nBlocks = BATCH * CHANS * STRIPES; // 16384
    EdgeAttention_fused_kernel<<<nBlocks, 256, 0, stream>>>(
        x, dww, gamma, beta, mean, var, out);
}